// ImputationLoss_17214228922848
// MI455X (gfx1250) — compile-verified
//
#include <hip/hip_runtime.h>
#include <hip/hip_bf16.h>

#define B_      32
#define L_      4096
#define C_      4
#define GROUP_  4
#define EPS_    1e-8f
#define JS_EPS_ 1e-7f
#define NSLICE  32
#define LSLICE  (L_ / NSLICE)   // 128
#define NPAIRS  496             // B*(B-1)/2, pairs with i < j

typedef __attribute__((ext_vector_type(2))) float v2f;
typedef __attribute__((ext_vector_type(8))) float v8f;

// Deterministic block tree-reduction (blockDim.x == 256).
__device__ __forceinline__ float block_reduce_sum(float v, float* sdata) {
    const int t = threadIdx.x;
    __syncthreads();                 // protect sdata from previous use
    sdata[t] = v;
    __syncthreads();
    for (int s = 128; s > 0; s >>= 1) {
        if (t < s) sdata[t] += sdata[t + s];
        __syncthreads();
    }
    return sdata[0];
}

// ---------------------------------------------------------------------------
// Kernel 1: per-(b,l) label, entropy term H = sum_c p*log(p+eps), CE partials.
// grid 512 x 256, one thread per (b,l).
// ---------------------------------------------------------------------------
__global__ void prep_kernel(const float* __restrict__ logits,
                            const float* __restrict__ prob,
                            const float* __restrict__ y_true,
                            float* __restrict__ H,
                            unsigned char* __restrict__ lab,
                            float* __restrict__ ce_partial) {
    __shared__ float sdata[256];
    const int idx = blockIdx.x * 256 + threadIdx.x;      // 0 .. B*L-1

    const float4 y = *(const float4*)(y_true + (size_t)idx * 4);
    int   lb = 0;
    float best = y.x;
    if (y.y > best) { best = y.y; lb = 1; }
    if (y.z > best) { best = y.z; lb = 2; }
    if (y.w > best) { best = y.w; lb = 3; }
    lab[idx] = (unsigned char)lb;

    const float4 p = *(const float4*)(prob + (size_t)idx * 4);
    H[idx] = p.x * __logf(p.x + JS_EPS_) + p.y * __logf(p.y + JS_EPS_)
           + p.z * __logf(p.z + JS_EPS_) + p.w * __logf(p.w + JS_EPS_);

    const float4 x = *(const float4*)(logits + (size_t)idx * 4);
    const float mx  = fmaxf(fmaxf(x.x, x.y), fmaxf(x.z, x.w));
    const float lse = __logf(__expf(x.x - mx) + __expf(x.y - mx) +
                             __expf(x.z - mx) + __expf(x.w - mx));
    const float xl  = (lb == 0) ? x.x : (lb == 1) ? x.y : (lb == 2) ? x.z : x.w;
    const float ce  = mx + lse - xl;     // = -logp[label]

    const float s = block_reduce_sum(ce, sdata);
    if (threadIdx.x == 0) ce_partial[blockIdx.x] = s;
}

// ---------------------------------------------------------------------------
// Kernel 2: R2 loss partials.  grid 128 x 256, one thread per (group, l).
// ---------------------------------------------------------------------------
__global__ void r2_kernel(const float* __restrict__ prob,
                          const unsigned char* __restrict__ lab,
                          float* __restrict__ r2_partial) {
    __shared__ float sdata[256];
    const int idx = blockIdx.x * 256 + threadIdx.x;      // 0 .. 8*4096-1
    const int g = idx >> 12;
    const int l = idx & (L_ - 1);

    int   cnt = 0;
    float pa[4];
#pragma unroll
    for (int u = 0; u < 4; ++u) {
        const int row = g * GROUP_ + u;
        cnt += (lab[row * L_ + l] != 0) ? 1 : 0;
        const float4 q = *(const float4*)(prob + ((size_t)row * L_ + l) * 4);
        pa[u] = q.y + q.z + q.w;
    }
    const bool  af_mask = (cnt == 0) || (cnt == GROUP_);
    const float af      = af_mask ? 0.5f : (float)cnt * 0.25f;
    float denom = af * (1.0f - af);
    denom = (denom < 0.01f) ? 0.01f : denom;

    float ss = 0.0f;
#pragma unroll
    for (int u = 0; u < 4; ++u) { const float d = pa[u] - af; ss += d * d; }
    const float r2v = af_mask ? 0.0f : (ss * 0.25f) / denom;

    const float s = block_reduce_sum(r2v, sdata);
    if (threadIdx.x == 0) r2_partial[blockIdx.x] = s;
}

// ---------------------------------------------------------------------------
// Kernel 3: cross term of JS, UPPER TRIANGLE ONLY (js is symmetric; the
// diagonal is exactly zero in the reference, handled in final_kernel):
//   jsx[i*32+j] = sum_l mask * sum_c (p_ic + q_jc) * log(0.5*(p+q) + eps)
// grid NPAIRS x 256, one block per unordered pair (i < j).
// ---------------------------------------------------------------------------
__global__ void cross_kernel(const float* __restrict__ prob,
                             const unsigned char* __restrict__ lab,
                             float* __restrict__ jsx) {
    __shared__ float sdata[256];
    // Map linear pair id -> (i, j) with i < j (uniform scalar loop, <= 31 iters).
    int pid = blockIdx.x;
    int i = 0;
    while (pid >= (B_ - 1 - i)) { pid -= (B_ - 1 - i); ++i; }
    const int j = i + 1 + pid;

    const float*         pi = prob + (size_t)i * L_ * 4;
    const float*         pj = prob + (size_t)j * L_ * 4;
    const unsigned char* li = lab + i * L_;
    const unsigned char* lj = lab + j * L_;

    float acc = 0.0f;
    for (int l = threadIdx.x; l < L_; l += 256) {
        const float4 p = *(const float4*)(pi + (size_t)l * 4);
        const float4 q = *(const float4*)(pj + (size_t)l * 4);
        const float mask = (li[l] == lj[l]) ? 1.0f : 0.0f;
        const float s0 = p.x + q.x, s1 = p.y + q.y;
        const float s2 = p.z + q.z, s3 = p.w + q.w;
        const float c = s0 * __logf(0.5f * s0 + JS_EPS_)
                      + s1 * __logf(0.5f * s1 + JS_EPS_)
                      + s2 * __logf(0.5f * s2 + JS_EPS_)
                      + s3 * __logf(0.5f * s3 + JS_EPS_);
        acc += mask * c;
    }
    const float s = block_reduce_sum(acc, sdata);
    if (threadIdx.x == 0) jsx[i * B_ + j] = s;
}

// ---------------------------------------------------------------------------
// Kernel 4: entropy term via WMMA f32 16x16x4.
//   S1[i,j] = sum_{l,k} (H_i(l)*y[i,l,k]) * y[j,l,k]   (32x32, K = 16384)
// One K=4 step == one sequence position l (C_==4).
// A-layout (ISA 16x4 f32): lane = m + 16*half, VGPR v -> K = v + 2*half.
// grid = 4 tiles * NSLICE slices, 32 threads (1 wave, EXEC all ones).
// ---------------------------------------------------------------------------
__global__ void ent_wmma_kernel(const float* __restrict__ y_true,
                                const float* __restrict__ H,
                                float* __restrict__ s1p) {
    const int tile  = blockIdx.x / NSLICE;   // 0..3 : (rowTile, colTile)
    const int slice = blockIdx.x % NSLICE;
    const int lane  = threadIdx.x;           // 0..31
    const int half  = lane >> 4;             // 0/1 -> K offset 0/2
    const int mn    = lane & 15;
    const int i0    = (tile >> 1) * 16;
    const int j0    = (tile & 1) * 16;
    const int rowA  = i0 + mn;
    const int rowB  = j0 + mn;

    const float* ya = y_true + ((size_t)rowA * L_) * 4 + 2 * half;
    const float* yb = y_true + ((size_t)rowB * L_) * 4 + 2 * half;
    const float* hA = H + (size_t)rowA * L_;

    v8f c = {0.f, 0.f, 0.f, 0.f, 0.f, 0.f, 0.f, 0.f};
    const int l0 = slice * LSLICE;
    for (int l = l0; l < l0 + LSLICE; ++l) {
        const float2 t1 = *(const float2*)(ya + (size_t)l * 4);
        const float2 t2 = *(const float2*)(yb + (size_t)l * 4);
        const float  h  = hA[l];
        v2f a, b;
        a[0] = h * t1.x;  a[1] = h * t1.y;
        b[0] = t2.x;      b[1] = t2.y;
        // 8 args: (neg_a, A, neg_b, B, c_mod, C, reuse_a, reuse_b)
        c = __builtin_amdgcn_wmma_f32_16x16x4_f32(false, a, false, b,
                                                  (short)0, c, false, false);
    }

    float* out = s1p + ((size_t)(tile * NSLICE + slice)) * 256 + lane * 8;
#pragma unroll
    for (int v = 0; v < 8; ++v) out[v] = c[v];
}

// ---------------------------------------------------------------------------
// Kernel 5: assemble S1 from slice partials (C/D layout: M = v + 8*(lane>=16),
// N = lane&15), build evo weights, combine all losses. 1 block x 256 threads.
// ---------------------------------------------------------------------------
__global__ void final_kernel(const float* __restrict__ y_evo,
                             const float* __restrict__ ce_partial,  // 512
                             const float* __restrict__ r2_partial,  // 128
                             const float* __restrict__ jsx,         // 1024 (upper tri valid)
                             const float* __restrict__ s1p,         // 4*NSLICE*256
                             float* __restrict__ out) {
    __shared__ float sdata[256];
    __shared__ float sS1[1024];
    __shared__ float srow[32];
    const int t = threadIdx.x;

    // Assemble S1[i*32+j] from WMMA slice partials (deterministic order).
    for (int e = t; e < 1024; e += 256) {
        const int tile = e >> 8;
        const int r    = e & 255;
        const int lane = r >> 3;
        const int v    = r & 7;
        const float* base = s1p + (size_t)tile * NSLICE * 256 + lane * 8 + v;
        float s = 0.0f;
        for (int sl = 0; sl < NSLICE; ++sl) s += base[(size_t)sl * 256];
        const int m = (v + ((lane >> 4) << 3)) + (tile >> 1) * 16;
        const int n = (lane & 15) + (tile & 1) * 16;
        sS1[m * 32 + n] = s;
    }
    __syncthreads();

    if (t < 32) {
        float rs = 0.0f;
        for (int j = 0; j < 32; ++j) rs += __expf(-y_evo[t * 32 + j]);
        srow[t] = rs + EPS_;
    }
    __syncthreads();

    float evo_acc = 0.0f;
    for (int e = t; e < 1024; e += 256) {
        const int i = e >> 5;
        const int j = e & 31;
        if (i != j) {                         // diagonal js is exactly 0
            const int lo = (i < j) ? i : j;
            const int hi = (i < j) ? j : i;
            const float w  = __expf(-y_evo[e]) / srow[i];
            const float js = 0.5f * (sS1[i * 32 + j] + sS1[j * 32 + i]
                                     - jsx[lo * B_ + hi]);
            evo_acc += w * js;
        }
    }
    const float evo = block_reduce_sum(evo_acc, sdata);
    const float ce  = block_reduce_sum(ce_partial[t] + ce_partial[t + 256], sdata);
    const float r2s = block_reduce_sum((t < 128) ? r2_partial[t] : 0.0f, sdata);

    if (t == 0) {
        const float r2_loss = -(r2s * (float)GROUP_) / (float)B_;  // R2_WEIGHT=1
        out[0] = ce + r2_loss + evo;   // total
        out[1] = ce;                   // ce_loss
        out[2] = r2_loss;              // r2_loss
        out[3] = evo;                  // evo_loss (EVO_WEIGHT=1)
    }
}

// ---------------------------------------------------------------------------
extern "C" void kernel_launch(void* const* d_in, const int* in_sizes, int n_in,
                              void* d_out, int out_size, void* d_ws, size_t ws_size,
                              hipStream_t stream) {
    const float* logits = (const float*)d_in[0];
    const float* prob   = (const float*)d_in[1];
    const float* y_true = (const float*)d_in[2];
    const float* y_evo  = (const float*)d_in[3];

    // Workspace layout (floats), ~790 KB total.
    float* H   = (float*)d_ws;                 // B*L           = 131072
    float* cep = H + (size_t)B_ * L_;          // 512
    float* r2p = cep + 512;                    // 128
    float* jsx = r2p + 128;                    // 1024 (upper triangle written)
    float* s1p = jsx + 1024;                   // 4*NSLICE*256  = 32768
    unsigned char* lab = (unsigned char*)(s1p + 4 * NSLICE * 256);  // B*L bytes

    prep_kernel<<<512, 256, 0, stream>>>(logits, prob, y_true, H, lab, cep);
    r2_kernel<<<128, 256, 0, stream>>>(prob, lab, r2p);
    cross_kernel<<<NPAIRS, 256, 0, stream>>>(prob, lab, jsx);
    ent_wmma_kernel<<<4 * NSLICE, 32, 0, stream>>>(y_true, H, s1p);
    final_kernel<<<1, 256, 0, stream>>>(y_evo, cep, r2p, jsx, s1p, (float*)d_out);
}